// TextGuide_69286412419152
// MI455X (gfx1250) — compile-verified
//
#include <hip/hip_runtime.h>
#include <hip/hip_bf16.h>
#include <math.h>

typedef float v2f __attribute__((ext_vector_type(2)));
typedef float v8f __attribute__((ext_vector_type(8)));

constexpr int kB      = 16;
constexpr int kC      = 256;
constexpr int kHW     = 1024;   // 32*32 pixels
constexpr int kHeads  = 4;
constexpr int kHD     = 64;     // head dim
constexpr int kGroups = 16;
constexpr int kCPG    = kC / kGroups;  // 16 channels per group
constexpr float kEPS  = 1e-5f;

// ---------------------------------------------------------------------------
// f32 dense WMMA: D(16x16,f32) = A(16x4,f32) * B(4x16,f32) + C
// A frag: lane<16 -> M=lane, K = {0,1}; lane>=16 -> M=lane-16, K = {2,3}
// B frag: lane<16 -> N=lane, K = {0,1}; lane>=16 -> N=lane-16, K = {2,3}
// C/D   : vgpr r, lane<16 -> (M=r, N=lane); lane>=16 -> (M=r+8, N=lane-16)
// ---------------------------------------------------------------------------
__device__ __forceinline__ v8f wmma4(v2f a, v2f b, v8f c) {
  return __builtin_amdgcn_wmma_f32_16x16x4_f32(
      /*neg_a=*/false, a, /*neg_b=*/false, b,
      /*c_mod=*/(short)0, c, /*reuse_a=*/false, /*reuse_b=*/false);
}

// gfx1250 async global->LDS copy (ASYNCcnt path). 16B per lane.
__device__ __forceinline__ void async_g2l_b128(unsigned lds_byte_ofs,
                                               const float* gaddr) {
  asm volatile("global_load_async_to_lds_b128 %0, %1, off"
               :: "v"(lds_byte_ofs), "v"(gaddr) : "memory");
}
__device__ __forceinline__ void wait_async0() {
  asm volatile("s_wait_asynccnt 0" ::: "memory");
}

// ---------------------------------------------------------------------------
// GroupNorm: one block per (batch, group); 16 ch * 1024 px = 16384 elements
// ---------------------------------------------------------------------------
__global__ __launch_bounds__(256) void groupnorm_kernel(
    const float* __restrict__ x, const float* __restrict__ gw,
    const float* __restrict__ gb, float* __restrict__ y) {
  const int bg = blockIdx.x;           // b*kGroups + g
  const int g  = bg % kGroups;
  const size_t base = (size_t)bg * kCPG * kHW;   // == b*kC*kHW + g*kCPG*kHW
  const float* xp = x + base;

  float s = 0.f, s2 = 0.f;
  for (int i = threadIdx.x; i < kCPG * kHW; i += blockDim.x) {
    float v = xp[i];
    s += v; s2 += v * v;
  }
  for (int off = 16; off > 0; off >>= 1) {
    s  += __shfl_down(s,  off, 32);
    s2 += __shfl_down(s2, off, 32);
  }
  __shared__ float rs[8], rs2[8];
  __shared__ float smu, srstd;
  const int wv = threadIdx.x >> 5, ln = threadIdx.x & 31;
  if (ln == 0) { rs[wv] = s; rs2[wv] = s2; }
  __syncthreads();
  if (threadIdx.x == 0) {
    float ts = 0.f, ts2 = 0.f;
    for (int i = 0; i < 8; ++i) { ts += rs[i]; ts2 += rs2[i]; }
    const float cnt = (float)(kCPG * kHW);
    float mu  = ts / cnt;
    float var = ts2 / cnt - mu * mu;
    smu = mu; srstd = rsqrtf(var + kEPS);
  }
  __syncthreads();
  const float mu = smu, rstd = srstd;
  for (int i = threadIdx.x; i < kCPG * kHW; i += blockDim.x) {
    int c = g * kCPG + i / kHW;
    y[base + i] = (xp[i] - mu) * rstd * gw[c] + gb[c];
  }
}

// ---------------------------------------------------------------------------
// QKV projection: Y[b][n][c][p] = sum_k W[n*192 + sel*64 + c][k] * X[b][k][p]
// sel: 0 = Q rows, 1 = K rows, 2 = V rows.  One wave -> 16(out ch) x 16(px).
// grid = (8, 16, B); block = 256 (8 waves, each a distinct pixel tile)
// ---------------------------------------------------------------------------
__global__ __launch_bounds__(256) void qkv_kernel(
    const float* __restrict__ W, const float* __restrict__ X,
    float* __restrict__ Y, int sel) {
  const int wave = threadIdx.x >> 5, lane = threadIdx.x & 31;
  const int half = lane >> 4, idx = lane & 15;
  const int p0 = blockIdx.x * 128 + wave * 16;
  const int n  = blockIdx.y >> 2;       // head
  const int ct = blockIdx.y & 3;        // 16-channel tile within head
  const int wrow0 = n * 192 + sel * 64 + ct * 16;
  const float* Wp = W + (size_t)wrow0 * kC;
  const float* Xp = X + (size_t)blockIdx.z * kC * kHW;

  v8f acc = {};
  for (int k0 = 0; k0 < kC; k0 += 4) {
    const int kk = k0 + 2 * half;
    v2f a;  a.x = Wp[idx * kC + kk];            a.y = Wp[idx * kC + kk + 1];
    v2f bf; bf.x = Xp[(size_t)kk * kHW + p0 + idx];
            bf.y = Xp[(size_t)(kk + 1) * kHW + p0 + idx];
    acc = wmma4(a, bf, acc);
  }
  float* Yp = Y + (((size_t)blockIdx.z * kHeads + n) * kHD + ct * 16 + 8 * half)
                  * (size_t)kHW + p0 + idx;
  #pragma unroll
  for (int r = 0; r < 8; ++r) Yp[(size_t)r * kHW] = acc[r];
}

// ---------------------------------------------------------------------------
// Flash attention per (b, head).  All 8 waves of a block share one (b, head);
// each 64x16 K/V tile is staged ONCE per block into LDS via async
// global->LDS b128 copies (coalesced 16B/lane), then all waves feed WMMA
// fragments from LDS.  Row stride 20 floats keeps b128 stores 16B-aligned
// and frag reads bank-conflict-free.  Per-wave P staged in LDS for the
// P*V^T A-fragments.  Output stored as Xatt[b][p][n*64+c].
// ---------------------------------------------------------------------------
__global__ __launch_bounds__(256) void attn_kernel(
    const float* __restrict__ Q, const float* __restrict__ K,
    const float* __restrict__ V, float* __restrict__ Xatt) {
  __shared__ float Kt[64][20];       // K tile: [c][yx], stride 20 (80B rows)
  __shared__ float Vt[64][20];       // V tile: [c][yx]
  __shared__ float Plds[8][16][17];  // per-wave P tile
  const int tid  = threadIdx.x;
  const int wave = tid >> 5, lane = tid & 31;
  const int half = lane >> 4, idx = lane & 15;
  const int wid = blockIdx.x * 8 + wave;     // 0..4095
  const int qt  = wid & 63;                  // query tile (per wave)
  const int bn  = wid >> 6;                  // b*4 + n (uniform in block)
  const int n   = bn & 3, b = bn >> 2;
  const float* Qp = Q + (size_t)bn * kHD * kHW;
  const float* Kp = K + (size_t)bn * kHD * kHW;
  const float* Vp = V + (size_t)bn * kHD * kHW;
  const int q0 = qt * 16;

  // staging assignment: thread -> (channel row sc, 4-float column chunk x4)
  const int sc = tid >> 2;            // 0..63
  const int x4 = (tid & 3) * 4;       // 0,4,8,12
  const unsigned kofs = (unsigned)(uintptr_t)&Kt[sc][x4];
  const unsigned vofs = (unsigned)(uintptr_t)&Vt[sc][x4];

  // preload A frags of Q^T (row = query px, K dim = channel 0..63)
  v2f aq[16];
  #pragma unroll
  for (int ks = 0; ks < 16; ++ks) {
    const int c = ks * 4 + 2 * half;
    aq[ks].x = Qp[(size_t)c * kHW + q0 + idx];
    aq[ks].y = Qp[(size_t)(c + 1) * kHW + q0 + idx];
  }

  v8f o0 = {}, o1 = {}, o2 = {}, o3 = {};
  float m[8], l[8];
  #pragma unroll
  for (int r = 0; r < 8; ++r) { m[r] = -3.0e38f; l[r] = 0.f; }
  const float inv_sqrt_c = 0.0625f;   // 1/sqrt(256)

  for (int j = 0; j < 64; ++j) {
    const int y0 = j * 16;
    __syncthreads();                            // LDS tile reuse fence
    async_g2l_b128(kofs, Kp + (size_t)sc * kHW + y0 + x4);
    async_g2l_b128(vofs, Vp + (size_t)sc * kHW + y0 + x4);
    wait_async0();                              // own async copies done
    __syncthreads();                            // tile visible to all waves

    // ---- S tile = Q^T K (K frags from LDS) ----
    v8f s = {};
    #pragma unroll
    for (int ks = 0; ks < 16; ++ks) {
      const int c = ks * 4 + 2 * half;
      v2f bk;
      bk.x = Kt[c][idx];
      bk.y = Kt[c + 1][idx];
      s = wmma4(aq[ks], bk, s);
    }
    // ---- online softmax (row stats within each 16-lane half) ----
    float sc8[8];
    #pragma unroll
    for (int r = 0; r < 8; ++r) {
      float sv = s[r] * inv_sqrt_c;
      float mx = sv;
      mx = fmaxf(mx, __shfl_xor(mx, 1, 32));
      mx = fmaxf(mx, __shfl_xor(mx, 2, 32));
      mx = fmaxf(mx, __shfl_xor(mx, 4, 32));
      mx = fmaxf(mx, __shfl_xor(mx, 8, 32));
      const float mnew = fmaxf(m[r], mx);
      const float p = __expf(sv - mnew);
      float rsum = p;
      rsum += __shfl_xor(rsum, 1, 32);
      rsum += __shfl_xor(rsum, 2, 32);
      rsum += __shfl_xor(rsum, 4, 32);
      rsum += __shfl_xor(rsum, 8, 32);
      sc8[r] = __expf(m[r] - mnew);
      l[r]   = l[r] * sc8[r] + rsum;
      m[r]   = mnew;
      Plds[wave][r + 8 * half][idx] = p;   // stage P for A-frag reload
    }
    #pragma unroll
    for (int r = 0; r < 8; ++r) {
      o0[r] *= sc8[r]; o1[r] *= sc8[r]; o2[r] *= sc8[r]; o3[r] *= sc8[r];
    }
    asm volatile("s_wait_dscnt 0" ::: "memory");   // same-wave LDS RAW on P

    // ---- O += P(16x16) * V^T(16x64) (V frags from LDS) ----
    #pragma unroll
    for (int ks = 0; ks < 4; ++ks) {
      const int kk = ks * 4 + 2 * half;
      v2f ap; ap.x = Plds[wave][idx][kk]; ap.y = Plds[wave][idx][kk + 1];
      v2f bv;
      bv.x = Vt[0 * 16 + idx][kk]; bv.y = Vt[0 * 16 + idx][kk + 1];
      o0 = wmma4(ap, bv, o0);
      bv.x = Vt[1 * 16 + idx][kk]; bv.y = Vt[1 * 16 + idx][kk + 1];
      o1 = wmma4(ap, bv, o1);
      bv.x = Vt[2 * 16 + idx][kk]; bv.y = Vt[2 * 16 + idx][kk + 1];
      o2 = wmma4(ap, bv, o2);
      bv.x = Vt[3 * 16 + idx][kk]; bv.y = Vt[3 * 16 + idx][kk + 1];
      o3 = wmma4(ap, bv, o3);
    }
  }

  // normalize and store as (b, p, c) with c = n*64 + ct*16 + idx (coalesced)
  #pragma unroll
  for (int r = 0; r < 8; ++r) {
    const float rcp = 1.0f / l[r];
    const int p = q0 + r + 8 * half;
    const size_t base = ((size_t)b * kHW + p) * kC + n * kHD;
    Xatt[base + 0 * 16 + idx] = o0[r] * rcp;
    Xatt[base + 1 * 16 + idx] = o1[r] * rcp;
    Xatt[base + 2 * 16 + idx] = o2[r] * rcp;
    Xatt[base + 3 * 16 + idx] = o3[r] * rcp;
  }
}

// ---------------------------------------------------------------------------
// Output projection + bias + residual:
// Y[b][o][p] = sum_c Wout[o][c] * Xatt[b][p][c] + bout[o] + textn[b][o][p]
// ---------------------------------------------------------------------------
__global__ __launch_bounds__(256) void proj_out_kernel(
    const float* __restrict__ Wout, const float* __restrict__ bout,
    const float* __restrict__ Xatt, const float* __restrict__ textn,
    float* __restrict__ Y) {
  const int wave = threadIdx.x >> 5, lane = threadIdx.x & 31;
  const int half = lane >> 4, idx = lane & 15;
  const int p0 = blockIdx.x * 128 + wave * 16;
  const int o0 = blockIdx.y * 16;
  const int b  = blockIdx.z;
  const float* Wp = Wout + (size_t)o0 * kC;
  const float* Xp = Xatt + (size_t)b * kHW * kC;

  v8f acc = {};
  for (int k0 = 0; k0 < kC; k0 += 4) {
    const int kk = k0 + 2 * half;
    v2f a;  a.x  = Wp[idx * kC + kk];           a.y  = Wp[idx * kC + kk + 1];
    v2f bv; bv.x = Xp[(size_t)(p0 + idx) * kC + kk];
            bv.y = Xp[(size_t)(p0 + idx) * kC + kk + 1];
    acc = wmma4(a, bv, acc);
  }
  #pragma unroll
  for (int r = 0; r < 8; ++r) {
    const int o = o0 + r + 8 * half;
    const size_t off = ((size_t)b * kC + o) * kHW + p0 + idx;
    Y[off] = acc[r] + bout[o] + textn[off];
  }
}

// ---------------------------------------------------------------------------
extern "C" void kernel_launch(void* const* d_in, const int* in_sizes, int n_in,
                              void* d_out, int out_size, void* d_ws, size_t ws_size,
                              hipStream_t stream) {
  const float* fea_img  = (const float*)d_in[0];
  const float* fea_text = (const float*)d_in[1];
  const float* gnA_w    = (const float*)d_in[2];
  const float* gnA_b    = (const float*)d_in[3];
  const float* gnB_w    = (const float*)d_in[4];
  const float* gnB_b    = (const float*)d_in[5];
  const float* W_qkv_A  = (const float*)d_in[6];
  const float* W_qkv_B  = (const float*)d_in[7];
  const float* W_out    = (const float*)d_in[8];
  const float* b_out    = (const float*)d_in[9];
  float* out = (float*)d_out;

  float* ws = (float*)d_ws;
  const size_t SEG = (size_t)kB * kC * kHW;   // 4,194,304 floats (16 MB)
  float* img_n  = ws + 0 * SEG;
  float* text_n = ws + 1 * SEG;
  float* Qb     = ws + 2 * SEG;   // (B, 4, 64, 1024)
  float* Kb     = ws + 3 * SEG;
  float* Vb     = ws + 4 * SEG;
  float* Xatt   = ws + 5 * SEG;   // (B, 1024, 256)

  groupnorm_kernel<<<kB * kGroups, 256, 0, stream>>>(fea_img,  gnA_w, gnA_b, img_n);
  groupnorm_kernel<<<kB * kGroups, 256, 0, stream>>>(fea_text, gnB_w, gnB_b, text_n);

  dim3 gq(8, 16, kB);
  qkv_kernel<<<gq, 256, 0, stream>>>(W_qkv_A, img_n,  Qb, 0);  // Q from img
  qkv_kernel<<<gq, 256, 0, stream>>>(W_qkv_B, text_n, Kb, 1);  // K from text
  qkv_kernel<<<gq, 256, 0, stream>>>(W_qkv_B, text_n, Vb, 2);  // V from text

  attn_kernel<<<512, 256, 0, stream>>>(Qb, Kb, Vb, Xatt);

  proj_out_kernel<<<gq, 256, 0, stream>>>(W_out, b_out, Xatt, text_n, out);
}